// VolumeRenderer_86715389706439
// MI455X (gfx1250) — compile-verified
//
#include <hip/hip_runtime.h>

#ifndef __has_builtin
#define __has_builtin(x) 0
#endif

#define L_SAMPLES 128
#define FAR_PAD 1.0e10f
#define EPS_T 1.0e-10f
#define WAVES_PER_BLOCK 8
#define RAYS_PER_BLOCK (WAVES_PER_BLOCK * 2)

// ---- CDNA5 async Global->LDS path (b128 DMAs; plain-load fallback) ----
#if __has_builtin(__builtin_amdgcn_global_load_async_to_lds_b128)
#define USE_ASYNC 1
#else
#define USE_ASYNC 0
#endif

#if __has_builtin(__builtin_amdgcn_s_wait_asynccnt)
#define WAIT_ASYNC(n) __builtin_amdgcn_s_wait_asynccnt(n)
#else
#define WAIT_ASYNC(n) asm volatile("s_wait_asynccnt " #n ::: "memory")
#endif

typedef __attribute__((ext_vector_type(4))) int v4i;
#define AS_GLOBAL_V4I(p) ((__attribute__((address_space(1))) v4i*)(void*)(p))
#define AS_LDS_V4I(p)    ((__attribute__((address_space(3))) v4i*)(void*)(p))

// v_mov_b32_dpp with old-preserve (bound_ctrl off): invalid source lanes keep `old_v`.
// Row ops are confined to 16-lane rows -> one ray per row, both scanned at once.
template <int CTRL>
__device__ __forceinline__ float dpp_mov_f(float old_v, float src) {
  return __builtin_bit_cast(float,
      __builtin_amdgcn_update_dpp(__builtin_bit_cast(int, old_v),
                                  __builtin_bit_cast(int, src),
                                  CTRL, 0xF, 0xF, false));
}

// Row-of-16 inclusive sum scan: full row sum lands in rlane 15 of each row.
__device__ __forceinline__ float row_sum_f(float v) {
  v += dpp_mov_f<0x111>(0.0f, v);   // row_shr:1
  v += dpp_mov_f<0x112>(0.0f, v);   // row_shr:2
  v += dpp_mov_f<0x114>(0.0f, v);   // row_shr:4
  v += dpp_mov_f<0x118>(0.0f, v);   // row_shr:8
  return v;
}

#if __has_builtin(__builtin_amdgcn_tanhf)
__device__ __forceinline__ float fast_sigmoid(float x) {
  return fmaf(__builtin_amdgcn_tanhf(x * 0.5f), 0.5f, 0.5f);   // v_tanh_f32
}
#else
__device__ __forceinline__ float fast_sigmoid(float x) {
  return 1.0f / (1.0f + __expf(-x));
}
#endif

// TWO rays per wave32 (ray = DPP row). Lane rl of each row owns blocked samples
// 8*rl..8*rl+7. 128-sample exclusive cumprod = serial local prefix + ONE shared
// row-scan (4 DPP muls) + exclusive shift. Reductions shared across rows too.
__global__ __launch_bounds__(256) void volrend_scan_kernel(
    const float* __restrict__ depth,   // [N,128,1]
    const float* __restrict__ rgb,     // [N,128,3]
    const float* __restrict__ sigma,   // [N,128,1]
    float* __restrict__ out,           // [N*3 color | N depth]
    int n_rays)
{
  const int lane = threadIdx.x & 31;
  const int wave = threadIdx.x >> 5;
  const int row  = lane >> 4;                 // which ray of the pair
  const int rl   = lane & 15;                 // lane within the ray's row
  const int ray0 = (blockIdx.x * WAVES_PER_BLOCK + wave) * 2;
  if (ray0 >= n_rays) return;                 // pairs complete (n_rays even)

  const long base = (long)ray0 * L_SAMPLES;   // sample base of the ray pair

  float dloc[9], sloc[8], rloc[24];

#if USE_ASYNC
  // Per-wave staging (floats): [0..255] depth, [256..511] sigma, [512..1279] rgb
  // for BOTH rays (adjacent in memory). Each lane moves 16B at lane*16; the
  // IOFFSET immediate (added to both addresses) selects the 512B window.
  __shared__ float lds[WAVES_PER_BLOCK][1280];
  {
    const float* gD = depth + base     + lane * 4;
    const float* gS = sigma + base     + lane * 4;
    const float* gR = rgb   + base * 3 + lane * 4;
    float* lD = &lds[wave][lane * 4];
    float* lS = &lds[wave][256 + lane * 4];
    float* lR = &lds[wave][512 + lane * 4];
    __builtin_amdgcn_global_load_async_to_lds_b128(AS_GLOBAL_V4I(gD), AS_LDS_V4I(lD), 0, 0);
    __builtin_amdgcn_global_load_async_to_lds_b128(AS_GLOBAL_V4I(gD), AS_LDS_V4I(lD), 512, 0);
    __builtin_amdgcn_global_load_async_to_lds_b128(AS_GLOBAL_V4I(gS), AS_LDS_V4I(lS), 0, 0);
    __builtin_amdgcn_global_load_async_to_lds_b128(AS_GLOBAL_V4I(gS), AS_LDS_V4I(lS), 512, 0);
    __builtin_amdgcn_global_load_async_to_lds_b128(AS_GLOBAL_V4I(gR), AS_LDS_V4I(lR), 0, 0);
    __builtin_amdgcn_global_load_async_to_lds_b128(AS_GLOBAL_V4I(gR), AS_LDS_V4I(lR), 512, 0);
    __builtin_amdgcn_global_load_async_to_lds_b128(AS_GLOBAL_V4I(gR), AS_LDS_V4I(lR), 1024, 0);
    __builtin_amdgcn_global_load_async_to_lds_b128(AS_GLOBAL_V4I(gR), AS_LDS_V4I(lR), 1536, 0);
    __builtin_amdgcn_global_load_async_to_lds_b128(AS_GLOBAL_V4I(gR), AS_LDS_V4I(lR), 2048, 0);
    __builtin_amdgcn_global_load_async_to_lds_b128(AS_GLOBAL_V4I(gR), AS_LDS_V4I(lR), 2560, 0);
  }
  WAIT_ASYNC(0);

  // Blocked 16B-aligned LDS reads (depth/sigma 32B per lane, rgb 96B per lane).
  {
    const int db = row * 128 + rl * 8;
    *(float4*)&dloc[0] = *(const float4*)&lds[wave][db + 0];
    *(float4*)&dloc[4] = *(const float4*)&lds[wave][db + 4];
    dloc[8] = lds[wave][db + 8];   // rl==15 reads next region: discarded via FAR_PAD
    const int sb = 256 + row * 128 + rl * 8;
    *(float4*)&sloc[0] = *(const float4*)&lds[wave][sb + 0];
    *(float4*)&sloc[4] = *(const float4*)&lds[wave][sb + 4];
    const int rb = 512 + row * 384 + rl * 24;
    #pragma unroll
    for (int k = 0; k < 6; ++k)
      *(float4*)&rloc[k * 4] = *(const float4*)&lds[wave][rb + k * 4];
  }
#else
  {
    const long db = base + row * 128 + rl * 8;
    *(float4*)&dloc[0] = *(const float4*)&depth[db + 0];
    *(float4*)&dloc[4] = *(const float4*)&depth[db + 4];
    dloc[8] = (rl < 15) ? depth[db + 8] : 0.0f;
    *(float4*)&sloc[0] = *(const float4*)&sigma[db + 0];
    *(float4*)&sloc[4] = *(const float4*)&sigma[db + 4];
    const long rb = (base + row * 128) * 3 + rl * 24;
    #pragma unroll
    for (int k = 0; k < 6; ++k)
      *(float4*)&rloc[k * 4] = *(const float4*)&rgb[rb + k * 4];
  }
#endif

  // ---- deltas / transmittance factors (sample 8*rl+j) ----
  float e[8], t[8];
  #pragma unroll
  for (int j = 0; j < 8; ++j) {
    float delta = dloc[j + 1] - dloc[j];
    if (j == 7) delta = (rl == 15) ? FAR_PAD : delta;   // global sample 127
    e[j] = __expf(-(fmaxf(sloc[j], 0.0f) * delta));
    t[j] = e[j] + EPS_T;                                // 1 - alpha + eps
  }

  // ---- per-lane product, shared row scan, exclusive transmittance ----
  const float q = ((t[0] * t[1]) * (t[2] * t[3])) * ((t[4] * t[5]) * (t[6] * t[7]));
  float p = q;                                 // row-confined inclusive scan
  p *= dpp_mov_f<0x111>(1.0f, p);
  p *= dpp_mov_f<0x112>(1.0f, p);
  p *= dpp_mov_f<0x114>(1.0f, p);
  p *= dpp_mov_f<0x118>(1.0f, p);
  float T = dpp_mov_f<0x111>(1.0f, p);         // exclusive: rlane 0 of each row -> 1

  // ---- weights + weighted accumulation (serial T chain: 2 muls/sample) ----
  float accR = 0.0f, accG = 0.0f, accB = 0.0f, accD = 0.0f;
  #pragma unroll
  for (int j = 0; j < 8; ++j) {
    const float w = (1.0f - e[j]) * T;
    T *= t[j];
    accR = fmaf(w, fast_sigmoid(rloc[j * 3 + 0]), accR);
    accG = fmaf(w, fast_sigmoid(rloc[j * 3 + 1]), accG);
    accB = fmaf(w, fast_sigmoid(rloc[j * 3 + 2]), accB);
    accD = fmaf(w, dloc[j], accD);
  }

  // ---- per-row reduction (shared DPP adds); row sums land in rlane 15 ----
  const float cR = row_sum_f(accR);
  const float cG = row_sum_f(accG);
  const float cB = row_sum_f(accB);
  const float cD = row_sum_f(accD);

  if (rl == 15) {
    const int ray = ray0 + row;
    out[(long)ray * 3 + 0] = cR;
    out[(long)ray * 3 + 1] = cG;
    out[(long)ray * 3 + 2] = cB;
    out[(long)n_rays * 3 + ray] = cD;
  }
}

extern "C" void kernel_launch(void* const* d_in, const int* in_sizes, int n_in,
                              void* d_out, int out_size, void* d_ws, size_t ws_size,
                              hipStream_t stream) {
  const float* depth = (const float*)d_in[0];
  const float* rgb   = (const float*)d_in[1];
  const float* sigma = (const float*)d_in[2];
  float* out = (float*)d_out;

  const int n_rays = in_sizes[0] / L_SAMPLES;    // depth is [N,128,1]
  const int blocks = (n_rays + RAYS_PER_BLOCK - 1) / RAYS_PER_BLOCK;

  volrend_scan_kernel<<<blocks, 32 * WAVES_PER_BLOCK, 0, stream>>>(
      depth, rgb, sigma, out, n_rays);
}